// AttentionMemory_81484119540549
// MI455X (gfx1250) — compile-verified
//
#include <hip/hip_runtime.h>
#include <math.h>
#include <stdint.h>

typedef float v2f __attribute__((ext_vector_type(2)));
typedef float v8f __attribute__((ext_vector_type(8)));

#define CKD 128          // channel depth (K of the GEMM)
#define NN  4096         // H*W = 64*64, both N_mem and N_query
#define MT  128          // M tile per block
#define QT  128          // Q tile per block
#define LDA 136          // padded LDS row stride (floats): 2*136 mod 64 = 16 -> halves hit disjoint banks

// ---------------------------------------------------------------------------
// Kernel 0: a_sq[b,m] = sum_c mk[b,c,m]^2
// ---------------------------------------------------------------------------
__global__ __launch_bounds__(256) void asq_kernel(const float* __restrict__ mk,
                                                  float* __restrict__ asq) {
    int idx = blockIdx.x * 256 + threadIdx.x;     // b*NN + m
    int b = idx >> 12;
    int m = idx & (NN - 1);
    const float* p = mk + (size_t)b * CKD * NN + m;
    float s = 0.f;
#pragma unroll 8
    for (int c = 0; c < CKD; ++c) {
        float v = p[(size_t)c * NN];
        s += v * v;
    }
    asq[idx] = s;
}

// ---------------------------------------------------------------------------
// Kernel 1: logits[b,m,q] = (2*sum_c mk[c,m]*qk[c,q] - asq[b,m]) / sqrt(CK)
// 128x128 tile per 8-wave block, full K=128 staged in LDS via async
// global->LDS copies (ASYNCcnt), f32 WMMA core.
// ---------------------------------------------------------------------------
__global__ __launch_bounds__(256) void gemm_logits_kernel(const float* __restrict__ mk,
                                                          const float* __restrict__ qk,
                                                          const float* __restrict__ asq,
                                                          float* __restrict__ out) {
    extern __shared__ float smem[];
    float* sA = smem;               // [CKD][LDA], A[m,c] stored as sA[c*LDA + m]
    float* sB = smem + CKD * LDA;   // [CKD][LDA], B[c,q] stored as sB[c*LDA + q]

    const int b     = blockIdx.z;
    const int mBase = blockIdx.y * MT;
    const int qBase = blockIdx.x * QT;
    const float* gA = mk + (size_t)b * CKD * NN;
    const float* gB = qk + (size_t)b * CKD * NN;
    const int tid = threadIdx.x;

    // Async global->LDS staging: 128 rows(c) x 32 x b128 per matrix.
    // Per lane: 16 bytes memory -> LDS, tracked by ASYNCcnt (no VGPR bounce).
    // Flat LDS-aperture addresses keep the LDS byte offset in the low 32 bits.
#pragma unroll
    for (int it = 0; it < 16; ++it) {
        int flat = tid + it * 256;
        int c  = flat >> 5;
        int x4 = (flat & 31) * 4;
        uint64_t ga = (uint64_t)(uintptr_t)(gA + (size_t)c * NN + mBase + x4);
        uint64_t gb = (uint64_t)(uintptr_t)(gB + (size_t)c * NN + qBase + x4);
        uint32_t la = (uint32_t)(uintptr_t)(sA + c * LDA + x4);
        uint32_t lb = (uint32_t)(uintptr_t)(sB + c * LDA + x4);
        asm volatile("global_load_async_to_lds_b128 %0, %1, off"
                     :: "v"(la), "v"(ga) : "memory");
        asm volatile("global_load_async_to_lds_b128 %0, %1, off"
                     :: "v"(lb), "v"(gb) : "memory");
    }
    asm volatile("s_wait_asynccnt 0x0" ::: "memory");
    __syncthreads();

    const int wave = tid >> 5;       // 0..7: which 16-row M strip
    const int lane = tid & 31;
    const int lh   = lane >> 4;      // half-wave select (K offset +2, M/N offset +8)
    const int ll   = lane & 15;
    const int m0   = wave * 16;

    v8f acc[8] = {};                 // 8 q-tiles of 16x16 f32 accumulators

#pragma unroll 2
    for (int k0 = 0; k0 < CKD; k0 += 4) {
        // A fragment 16x4: lanes 0-15 K=k0..k0+1, lanes 16-31 K=k0+2..k0+3
        const float* pa = sA + (k0 + 2 * lh) * LDA + m0 + ll;
        v2f a;
        a.x = pa[0];
        a.y = pa[LDA];
#pragma unroll
        for (int j = 0; j < 8; ++j) {
            const float* pb = sB + (k0 + 2 * lh) * LDA + j * 16 + ll;
            v2f bf;
            bf.x = pb[0];
            bf.y = pb[LDA];
            acc[j] = __builtin_amdgcn_wmma_f32_16x16x4_f32(
                /*neg_a=*/false, a, /*neg_b=*/false, bf,
                /*c_mod=*/(short)0, acc[j],
                /*reuse_a=*/false, /*reuse_b=*/false);
        }
    }

    // Epilogue: apply (2*ab - a_sq[m]) * rsqrt(CK) and stream logits out.
    const float scale = 0.08838834764831845f;   // 1/sqrt(128)
    float aq[8];
#pragma unroll
    for (int v = 0; v < 8; ++v)
        aq[v] = asq[b * NN + mBase + m0 + v + lh * 8];

    float* po = out + (size_t)b * NN * NN + (size_t)(mBase + m0) * NN + qBase;
#pragma unroll
    for (int j = 0; j < 8; ++j) {
#pragma unroll
        for (int v = 0; v < 8; ++v) {
            int mrow = v + lh * 8;                 // C/D layout: VGPR v -> M=v (+8 for hi half)
            float val = (2.f * acc[j][v] - aq[v]) * scale;
            po[(size_t)mrow * NN + j * 16 + ll] = val;
        }
    }
}

// ---------------------------------------------------------------------------
// Kernel 2: in-place softmax over the memory axis m (column-wise), one thread
// per (b, q) column; adjacent threads read adjacent q -> coalesced.
// ---------------------------------------------------------------------------
__global__ __launch_bounds__(256) void softmax_kernel(float* __restrict__ L) {
    size_t col = (size_t)blockIdx.x * 256 + threadIdx.x;   // b*NN + q
    size_t b = col >> 12;
    size_t q = col & (NN - 1);
    float* p = L + b * (size_t)NN * NN + q;

    float mmax = -3.402823466e38f;
    float ssum = 0.f;
    for (int m = 0; m < NN; ++m) {
        float x  = p[(size_t)m * NN];
        float nm = fmaxf(mmax, x);
        ssum = ssum * __expf(mmax - nm) + __expf(x - nm);
        mmax = nm;
    }
    float inv = 1.f / ssum;
    for (int m = 0; m < NN; ++m) {
        size_t off = (size_t)m * NN;
        p[off] = __expf(p[off] - mmax) * inv;
    }
}

// ---------------------------------------------------------------------------
extern "C" void kernel_launch(void* const* d_in, const int* in_sizes, int n_in,
                              void* d_out, int out_size, void* d_ws, size_t ws_size,
                              hipStream_t stream) {
    const float* Mk = (const float*)d_in[0];
    const float* Qk = (const float*)d_in[1];
    float* out = (float*)d_out;
    float* asq = (float*)d_ws;                      // 4*4096 floats = 64 KB

    const int B = 4;
    size_t smem_bytes = (size_t)2 * CKD * LDA * sizeof(float);   // ~136 KB (<=320 KB/WGP)
    hipFuncSetAttribute((const void*)gemm_logits_kernel,
                        hipFuncAttributeMaxDynamicSharedMemorySize,
                        (int)smem_bytes);

    asq_kernel<<<(B * NN) / 256, 256, 0, stream>>>(Mk, asq);

    dim3 grid(NN / QT, NN / MT, B);                 // 32 x 32 x 4 blocks
    gemm_logits_kernel<<<grid, 256, smem_bytes, stream>>>(Mk, Qk, asq, out);

    softmax_kernel<<<(B * NN) / 256, 256, 0, stream>>>(out);
}